// Attention_sample_relative_21225728377276
// MI455X (gfx1250) — compile-verified
//
#include <hip/hip_runtime.h>

typedef __attribute__((ext_vector_type(2))) float v2f;
typedef __attribute__((ext_vector_type(4))) float v4f;
typedef __attribute__((ext_vector_type(8))) float v8f;

#define BB    2
#define NH    4
#define DH    64
#define HH    64
#define WW    64
#define NPIX  (HH * WW)      // 4096
#define CC    256
#define GG    (BB * NH)      // 8
#define GNT   (GG * NPIX)    // 32768
#define MROWS (BB * NPIX)    // 8192
#define KDIM  256
#define KP    9
#define LDK   260            // padded LDS row (260 % 64 = 4 -> no bank conflicts)

__device__ __forceinline__ void wait_asynccnt0() {
#if __has_builtin(__builtin_amdgcn_s_wait_asynccnt)
    __builtin_amdgcn_s_wait_asynccnt(0);
#else
    asm volatile("s_wait_asynccnt 0x0" ::: "memory");
#endif
}

// ---------------------------------------------------------------------------
// Generic fp32 WMMA GEMM: Y[M, Ntot] = A[M, 256] * Wt[Ntot, 256]^T (+ bias)
// One 16-row M-stripe per block; the A tile is staged into LDS with the
// gfx1250 async global->LDS DMA path (ASYNCcnt), then 8 waves sweep N tiles
// with V_WMMA_F32_16X16X4_F32 (fp32 A/B, f32 accumulate). ISA layouts:
//   A frag (lane): rows lane&15, K = k0 + 2*(lane>>4) .. +1   -> v2f
//   B frag (lane): B[k, n] = Wt[n, k], same K split, n = n0+(lane&15)
//   C/D: VGPR r <-> row r (lanes 0-15) / row r+8 (lanes 16-31), col n0+(lane&15)
// ---------------------------------------------------------------------------
__global__ __launch_bounds__(256) void gemm_f32_wmma(const float* __restrict__ A,
                                                     const float* __restrict__ Wt,
                                                     const float* __restrict__ bias,
                                                     float* __restrict__ Y,
                                                     int Ntot)
{
    __shared__ float smem[16 * LDK];

    const int m0 = blockIdx.x << 4;
    // Async copy of the contiguous 16x256 A tile into padded LDS:
    // 256 threads x 4 iterations x 16B = 16 KB, tracked by ASYNCcnt.
    const float* Atile = A + (size_t)m0 * KDIM;
    for (int i = threadIdx.x * 4; i < 16 * KDIM; i += 256 * 4) {
        const int row = i >> 8, k = i & 255;
        // Low 32 bits of a generic LDS pointer are the LDS byte address.
        const unsigned int ldsAddr =
            (unsigned int)(size_t)(&smem[row * LDK + k]);
        const unsigned long long gAddr =
            (unsigned long long)(size_t)(&Atile[i]);
        asm volatile("global_load_async_to_lds_b128 %0, %1, off"
                     :: "v"(ldsAddr), "v"(gAddr) : "memory");
    }
    wait_asynccnt0();
    __syncthreads();

    const int wave = threadIdx.x >> 5;
    const int lane = threadIdx.x & 31;
    const int lr   = lane & 15;            // frag row / output column-in-tile
    const int koff = (lane >> 4) << 1;     // 0 for lanes 0-15, 2 for 16-31
    const int nTiles = Ntot >> 4;

    for (int nt = wave; nt < nTiles; nt += 8) {
        const int n0 = nt << 4;
        const float* wrow = Wt + (size_t)(n0 + lr) * KDIM + koff;
        const float* arow = &smem[lr * LDK + koff];

        // Prefetch this wave's next N-tile weight row (gfx1250 global_prefetch).
        if (nt + 8 < nTiles)
            __builtin_prefetch(Wt + (size_t)(n0 + 128 + lr) * KDIM, 0, 1);

        v8f acc = {0.f, 0.f, 0.f, 0.f, 0.f, 0.f, 0.f, 0.f};
#pragma unroll 4
        for (int k0 = 0; k0 < KDIM; k0 += 4) {
            const v2f af = *(const v2f*)(arow + k0);
            const v2f bf = *(const v2f*)(wrow + k0);
            acc = __builtin_amdgcn_wmma_f32_16x16x4_f32(
                      false, af, false, bf, (short)0, acc, false, false);
        }

        const int col = n0 + lr;
        const float bv = bias ? bias[col] : 0.0f;
        const int mbase = m0 + ((lane < 16) ? 0 : 8);
#pragma unroll
        for (int r = 0; r < 8; ++r)
            Y[(size_t)(mbase + r) * Ntot + col] = acc[r] + bv;
    }
}

// ---------------------------------------------------------------------------
// Offset projection: offs[g*N+p, j] = x[b, p, h*64 .. +63] . conv_offset_w[j, :]
// ---------------------------------------------------------------------------
__global__ __launch_bounds__(256) void offset_proj(const float* __restrict__ x,
                                                   const float* __restrict__ cw,
                                                   float* __restrict__ offs)
{
    const int tid = blockIdx.x * 256 + threadIdx.x;
    if (tid >= GNT * 18) return;
    const int j  = tid % 18;
    const int gp = tid / 18;
    const int g  = gp / NPIX, p = gp % NPIX;
    const int b  = g / NH,    h = g % NH;

    const float* xr = x + (size_t)(b * NPIX + p) * CC + h * DH;
    const float* wr = cw + j * DH;
    float acc = 0.f;
#pragma unroll 8
    for (int c = 0; c < DH; ++c) acc = fmaf(xr[c], wr[c], acc);
    offs[(size_t)gp * 18 + j] = acc;
}

// ---------------------------------------------------------------------------
// Fused deformable attention. One wave per (g, pixel); lane owns 2 channels.
// Samples K and V straight from the qkv buffer (layout [b*N+p][768]:
// q at col h*64, k at 256+h*64, v at 512+h*64), adds relative-position
// logits (reference's H==W mixing kept faithfully), softmax over 9 taps,
// writes pre-projection [B, N, C].
// ---------------------------------------------------------------------------
__global__ __launch_bounds__(256) void deform_attn(const float* __restrict__ qkv,
                                                   const float* __restrict__ offs,
                                                   const float* __restrict__ relh,
                                                   const float* __restrict__ relw,
                                                   float* __restrict__ hbuf)
{
    const int wid  = (blockIdx.x * 256 + threadIdx.x) >> 5;  // (g, p) id
    const int lane = threadIdx.x & 31;
    const int g = wid >> 12;            // / 4096
    const int p = wid & (NPIX - 1);
    const int b = g >> 2, h = g & 3;
    const int y = p >> 6, x = p & 63;
    const int c0 = lane << 1;

    const v2f q2  = *(const v2f*)(qkv + (size_t)(b * NPIX + p) * 768 + h * DH + c0);
    const int pT  = (x << 6) + y;       // transposed pixel for h_logits term
    const v2f qT2 = *(const v2f*)(qkv + (size_t)(b * NPIX + pT) * 768 + h * DH + c0);

    const float* kbase = qkv + (size_t)b * NPIX * 768 + 256 + h * DH + c0;
    const float* vbase = kbase + 256;

    // One coalesced load of the wave's 18 offsets (lanes 0..17), then
    // broadcast per tap with wave32 shuffles instead of 18 redundant loads.
    const float* orow = offs + (size_t)wid * 18;
    const float ov = (lane < 18) ? orow[lane] : 0.f;

    float logit[KP], va0[KP], va1[KP];
    const float scale = 0.125f;  // dh^-0.5, dh = 64

#pragma unroll
    for (int j = 0; j < KP; ++j) {
        const float dy = __shfl(ov, 2 * j, 32);
        const float dx = __shfl(ov, 2 * j + 1, 32);
        const float sy = (float)y + (float)(j / 3 - 1) + dy;
        const float sx = (float)x + (float)(j % 3 - 1) + dx;
        const float fy = floorf(sy), fx = floorf(sx);
        const float wy = sy - fy,   wx = sx - fx;
        const int iy0 = (int)fy, ix0 = (int)fx;

        const float cw4[4] = { (1.f - wy) * (1.f - wx), (1.f - wy) * wx,
                               wy * (1.f - wx),          wy * wx };
        const int cy[4] = { iy0, iy0, iy0 + 1, iy0 + 1 };
        const int cx[4] = { ix0, ix0 + 1, ix0, ix0 + 1 };

        float k0a = 0.f, k1a = 0.f, v0a = 0.f, v1a = 0.f;
#pragma unroll
        for (int t = 0; t < 4; ++t) {
            // wave-uniform branch: offsets identical across lanes of this wave
            if (cy[t] >= 0 && cy[t] < HH && cx[t] >= 0 && cx[t] < WW) {
                const size_t pix = (size_t)(cy[t] * WW + cx[t]) * 768;
                const v2f kk = *(const v2f*)(kbase + pix);
                const v2f vv = *(const v2f*)(vbase + pix);
                const float w = cw4[t];
                k0a = fmaf(w, kk.x, k0a); k1a = fmaf(w, kk.y, k1a);
                v0a = fmaf(w, vv.x, v0a); v1a = fmaf(w, vv.y, v1a);
            }
        }

        const int r = 63 + j - x;                    // always in [0, 71]
        const v2f rw2 = *(const v2f*)(relw + (size_t)r * DH + c0);
        const v2f rh2 = *(const v2f*)(relh + (size_t)r * DH + c0);

        float lg = (q2.x * k0a + q2.y * k1a) * scale;
        lg += q2.x  * rw2.x + q2.y  * rw2.y;
        lg += qT2.x * rh2.x + qT2.y * rh2.y;
        logit[j] = lg;
        va0[j] = v0a; va1[j] = v1a;
    }

    // Reduce each of the 9 logits across the 32 lanes (each lane held 2 ch).
#pragma unroll
    for (int j = 0; j < KP; ++j) {
        float v = logit[j];
        for (int m = 16; m > 0; m >>= 1) v += __shfl_xor(v, m, 32);
        logit[j] = v;
    }

    // Softmax over the 9 taps (replicated per lane — values identical).
    float mx = logit[0];
#pragma unroll
    for (int j = 1; j < KP; ++j) mx = fmaxf(mx, logit[j]);
    float s = 0.f, wsm[KP];
#pragma unroll
    for (int j = 0; j < KP; ++j) { wsm[j] = __expf(logit[j] - mx); s += wsm[j]; }
    const float inv = 1.0f / s;

    float o0 = 0.f, o1 = 0.f;
#pragma unroll
    for (int j = 0; j < KP; ++j) {
        const float w = wsm[j] * inv;
        o0 = fmaf(w, va0[j], o0);
        o1 = fmaf(w, va1[j], o1);
    }
    v2f o; o.x = o0; o.y = o1;
    *(v2f*)(hbuf + (size_t)(b * NPIX + p) * CC + h * DH + c0) = o;
}

// ---------------------------------------------------------------------------
// Inputs (setup_inputs order): x, qkv_w, conv_offset_w, proj_w, proj_b,
// rel_height, rel_width, H, W.  H/W are device scalars (==64, hardcoded).
// ---------------------------------------------------------------------------
extern "C" void kernel_launch(void* const* d_in, const int* in_sizes, int n_in,
                              void* d_out, int out_size, void* d_ws, size_t ws_size,
                              hipStream_t stream)
{
    (void)in_sizes; (void)n_in; (void)out_size; (void)ws_size;

    const float* x    = (const float*)d_in[0];
    const float* qkvw = (const float*)d_in[1];
    const float* cw   = (const float*)d_in[2];
    const float* pw   = (const float*)d_in[3];
    const float* pb   = (const float*)d_in[4];
    const float* relh = (const float*)d_in[5];
    const float* relw = (const float*)d_in[6];
    float* out = (float*)d_out;

    float* qkvbuf = (float*)d_ws;                          // 8192*768 f32 (25.2 MB)
    float* offs   = qkvbuf + (size_t)MROWS * 768;          // 32768*18  f32 ( 2.4 MB)
    float* hbuf   = offs   + (size_t)GNT * 18;             // 8192*256  f32 ( 8.0 MB)

    // 1) QKV projection (WMMA fp32, async LDS staging)
    gemm_f32_wmma<<<MROWS / 16, 256, 0, stream>>>(x, qkvw, nullptr, qkvbuf, 768);
    // 2) Deform offsets
    offset_proj<<<(GNT * 18 + 255) / 256, 256, 0, stream>>>(x, cw, offs);
    // 3) Fused gather + rel-pos + softmax + V-combine
    deform_attn<<<GNT / 8, 256, 0, stream>>>(qkvbuf, offs, relh, relw, hbuf);
    // 4) Output projection + bias (WMMA fp32)
    gemm_f32_wmma<<<MROWS / 16, 256, 0, stream>>>(hbuf, pw, pb, out, 256);
}